// HybridQNN_59287728554368
// MI455X (gfx1250) — compile-verified
//
#include <hip/hip_runtime.h>

// CDNA5 / gfx1250: wave32, WMMA fp32 16x16x4.
typedef __attribute__((ext_vector_type(2))) float v2f;
typedef __attribute__((ext_vector_type(8))) float v8f;

#define N_LAYERS 7
#define N_QUBITS 8
#define DIM 256  // 2^N_QUBITS

// ---------------------------------------------------------------------------
// Kernel A: build the circuit operator U (256x256, orthogonal) and store it
// in d_ws PRE-SWIZZLED into WMMA B-fragment order:
//
//   bp[((nt*64 + kk)*32 + lane)] = ( U[n][k], U[n][k+1] )   as float2,
//   lane = hf*16 + l16,  n = nt*16 + l16,  k = kk*4 + 2*hf
//
// so kernel B's B-fragment load is ONE fully-coalesced global_load_b64 per
// WMMA (256 contiguous bytes per wave). Total 256 KB -> L2-resident.
//
// Block j evolves basis column e_j through the circuit in LDS (col[i]=U[i][j]).
// Wire q acts on bit (7-q) of the flat amplitude index (wire 0 = MSB).
// ---------------------------------------------------------------------------
__global__ __launch_bounds__(256) void build_u_kernel(const float* __restrict__ w,
                                                      float* __restrict__ bp) {
  __shared__ float col[DIM];
  const int i = threadIdx.x;   // amplitude index (row of U)
  const int j = blockIdx.x;    // basis column (col of U)
  col[i] = (i == j) ? 1.0f : 0.0f;
  __syncthreads();
  for (int l = 0; l < N_LAYERS; ++l) {
    // RY(theta) on each wire: new0 = c*a0 - s*a1 ; new1 = s*a0 + c*a1
    for (int q = 0; q < N_QUBITS; ++q) {
      const float th = 0.5f * w[l * N_QUBITS + q];
      const float c = cosf(th);
      const float s = sinf(th);
      const int mask = 1 << (7 - q);
      const float a0 = col[i & ~mask];
      const float a1 = col[i | mask];
      __syncthreads();
      col[i] = (i & mask) ? (s * a0 + c * a1) : (c * a0 - s * a1);
      __syncthreads();
    }
    // CNOT(control=q, target=q+1): where control bit set, flip target bit.
    for (int q = 0; q < N_QUBITS - 1; ++q) {
      const int mc = 1 << (7 - q);
      const int mt = 1 << (6 - q);
      const int src = (i & mc) ? (i ^ mt) : i;
      const float v = col[src];
      __syncthreads();
      col[i] = v;
      __syncthreads();
    }
  }
  // col[i] = U[i][j]. Column j is component (j&1) of the pair covering
  // k = j&~1:  kk = k>>2,  hf = (k>>1)&1.  Row i = n:  nt = i>>4, l16 = i&15.
  const int k   = j & ~1;
  const int c   = j & 1;
  const int kk  = k >> 2;
  const int hf  = (k >> 1) & 1;
  const int nt  = i >> 4;
  const int l16 = i & 15;
  const int lane = hf * 16 + l16;
  bp[(size_t)(((nt * 64 + kk) * 32) + lane) * 2 + c] = col[i];
}

// ---------------------------------------------------------------------------
// Kernel B: fused  Y = X * U^T  (WMMA fp32 16x16x4, K=256),
// z = (P+ - P-)/(P+ + P-)  (orthogonality folds the normalization exactly),
// then the 16-unit relu MLP + sigmoid.
//
// 256 threads = 8 waves; each wave owns 16 batch rows; 512 blocks cover B.
// Data movement on MI455X:
//  - A operand (16 rows x K=256) = 64 x v2f = 128 VGPRs/lane: x is streamed
//    from HBM exactly once (64 MB ~= 2.8 us at 23.3 TB/s), register-resident
//    across all 16 N-tiles.
//  - B operand comes pre-swizzled in fragment order: one coalesced
//    global_load_b64 per WMMA from the 256 KB L2-resident pack.
// ---------------------------------------------------------------------------
__global__ __launch_bounds__(256) void qnn_fused_kernel(
    const float* __restrict__ x, const v2f* __restrict__ bp,
    const float* __restrict__ W1, const float* __restrict__ b1,
    const float* __restrict__ W2, const float* __restrict__ b2,
    float* __restrict__ out) {
  __shared__ float redP[8][16][16];   // [wave][row-in-tile][lane-partial]
  __shared__ float redM[8][16][16];

  const int lane = threadIdx.x & 31;
  const int wave = threadIdx.x >> 5;
  const int hf   = lane >> 4;         // lane half: K-offset selector
  const int l16  = lane & 15;
  const int rbase = blockIdx.x * 128 + wave * 16;

  // A-fragment (16x4, M x K): lane<16 holds M=l16, K={k0,k0+1};
  //                           lane>=16 holds M=l16, K={k0+2,k0+3}.
  const float* xrow = x + (size_t)(rbase + l16) * DIM + 2 * hf;

  // Register-resident A: one streaming pass over x.
  v2f afrag[64];
#pragma unroll
  for (int kk = 0; kk < 64; ++kk) {
    afrag[kk] = *(const v2f*)(xrow + 4 * kk);
  }

  float accP[8], accM[8];
#pragma unroll
  for (int v = 0; v < 8; ++v) { accP[v] = 0.f; accM[v] = 0.f; }

  for (int nt = 0; nt < 16; ++nt) {         // N-tiles; nt<8 -> +1, nt>=8 -> -1
    const v2f* btile = bp + (size_t)nt * 64 * 32 + lane;
    v8f acc = {0.f, 0.f, 0.f, 0.f, 0.f, 0.f, 0.f, 0.f};
#pragma unroll
    for (int kk = 0; kk < 64; ++kk) {
      const v2f b = btile[kk * 32];         // one coalesced b64 per WMMA
      // D = A(16x4) * B(4x16) + C   (fp32 WMMA)
      acc = __builtin_amdgcn_wmma_f32_16x16x4_f32(
          /*neg_a=*/false, afrag[kk], /*neg_b=*/false, b,
          /*c_mod=*/(short)0, acc, /*reuse_a=*/false, /*reuse_b=*/false);
    }
    if (nt < 8) {
#pragma unroll
      for (int v = 0; v < 8; ++v) accP[v] += acc[v] * acc[v];
    } else {
#pragma unroll
      for (int v = 0; v < 8; ++v) accM[v] += acc[v] * acc[v];
    }
  }

  // C/D layout: VGPR v -> row v (lanes 0-15) or row v+8 (lanes 16-31).
#pragma unroll
  for (int v = 0; v < 8; ++v) {
    redP[wave][hf * 8 + v][l16] = accP[v];
    redM[wave][hf * 8 + v][l16] = accM[v];
  }
  __syncthreads();

  if (threadIdx.x < 128) {                  // one thread per row of this block
    const int wv = threadIdx.x >> 4;
    const int r  = threadIdx.x & 15;
    float P = 0.f, M = 0.f;
#pragma unroll
    for (int i = 0; i < 16; ++i) { P += redP[wv][r][i]; M += redM[wv][r][i]; }
    const float z = (P - M) / (P + M);      // exact: U is orthogonal
    float t = b2[0];
#pragma unroll
    for (int jj = 0; jj < 16; ++jj) {
      float h = z * W1[jj] + b1[jj];        // W1 is (16,1), W2 is (1,16)
      h = h > 0.f ? h : 0.f;
      t += h * W2[jj];
    }
    out[(size_t)blockIdx.x * 128 + wv * 16 + r] = 1.f / (1.f + expf(-t));
  }
}

// ---------------------------------------------------------------------------
extern "C" void kernel_launch(void* const* d_in, const int* in_sizes, int n_in,
                              void* d_out, int out_size, void* d_ws, size_t ws_size,
                              hipStream_t stream) {
  const float* x  = (const float*)d_in[0];
  const float* w  = (const float*)d_in[1];
  const float* W1 = (const float*)d_in[2];
  const float* b1 = (const float*)d_in[3];
  const float* W2 = (const float*)d_in[4];
  const float* b2 = (const float*)d_in[5];
  float* out = (float*)d_out;
  float* bp  = (float*)d_ws;                 // 256*256 fp32 = 256 KB scratch

  const int B = in_sizes[0] / DIM;           // 65536
  build_u_kernel<<<DIM, DIM, 0, stream>>>(w, bp);
  qnn_fused_kernel<<<B / 128, 256, 0, stream>>>(x, (const v2f*)bp, W1, b1, W2, b2, out);
}